// JumpReLU_27023934226963
// MI455X (gfx1250) — compile-verified
//
#include <hip/hip_runtime.h>
#include <math.h>

// JumpReLU + per-row top-K for MI455X (gfx1250, wave32).
// One workgroup (256 thr = 8 wave32) per row:
//   1) Async-DMA the 64KB row into LDS via global_load_async_to_lds_b128
//      (ASYNCcnt-tracked, no VGPR staging) -> single HBM read pass.
//   2) Row max: LDS b128 strided scan + wave32 shuffle + cross-wave LDS.
//   3) thr = (max < 2) ? max - 1e-6 : 2 ; compact actives via ds_add_rtn.
//   4) 1024-entry bitonic sort in LDS (descending), emit K (value,index).
// Memory-bound: 256 MiB @ 23.3 TB/s ~= 12 us floor.

#define NF     16384
#define TPB    256
#define CAP    1024
#define CHUNKS (NF / 4 / TPB)   // 16 float4 loads per thread

typedef int v4i __attribute__((ext_vector_type(4)));
typedef __attribute__((address_space(1))) v4i gbl_v4i;   // global
typedef __attribute__((address_space(3))) v4i lds_v4i;   // LDS

#if defined(__gfx1250__) && __has_builtin(__builtin_amdgcn_global_load_async_to_lds_b128)
#define USE_ASYNC_LDS 1
#else
#define USE_ASYNC_LDS 0
#endif

__global__ __launch_bounds__(TPB) void jumprelu_topk_kernel(
    const float* __restrict__ x,
    float* __restrict__ out_val,
    int*   __restrict__ out_idx,
    int K)
{
    __shared__ __align__(16) float s_row[NF];   // 64 KB row stage
    __shared__ float s_val[CAP];                // 4 KB sort values
    __shared__ int   s_idx[CAP];                // 4 KB sort indices
    __shared__ float s_red[8];                  // per-wave maxima
    __shared__ int   s_cnt;

    const int tid = threadIdx.x;
    const int row = blockIdx.x;

    const float4* grow = (const float4*)(x + (size_t)row * NF);
    float4*       lrow = (float4*)s_row;

    if (tid == 0) s_cnt = 0;

    // ---- Stage 1: global -> LDS, one pass over HBM -----------------------
#if USE_ASYNC_LDS
    #pragma unroll
    for (int it = 0; it < CHUNKS; ++it) {
        const int i4 = tid + it * TPB;
        __builtin_amdgcn_global_load_async_to_lds_b128(
            (gbl_v4i*)(v4i*)(grow + i4),
            (lds_v4i*)(lrow + i4),
            /*imm offset*/0, /*cpol*/0);
    }
  #if __has_builtin(__builtin_amdgcn_s_wait_asynccnt)
    __builtin_amdgcn_s_wait_asynccnt(0);
  #else
    asm volatile("s_wait_asynccnt 0" ::: "memory");
  #endif
#else
    #pragma unroll
    for (int it = 0; it < CHUNKS; ++it) {
        const int i4 = tid + it * TPB;
        lrow[i4] = grow[i4];
    }
#endif
    __syncthreads();

    // ---- Stage 2: row max ------------------------------------------------
    float m = -INFINITY;
    #pragma unroll
    for (int it = 0; it < CHUNKS; ++it) {
        const float4 v = lrow[tid + it * TPB];
        m = fmaxf(m, fmaxf(fmaxf(v.x, v.y), fmaxf(v.z, v.w)));
    }
    // wave32 butterfly reduction
    #pragma unroll
    for (int off = 16; off >= 1; off >>= 1)
        m = fmaxf(m, __shfl_xor(m, off, 32));
    if ((tid & 31) == 0) s_red[tid >> 5] = m;
    __syncthreads();
    if (tid == 0) {
        float mm = s_red[0];
        #pragma unroll
        for (int w = 1; w < 8; ++w) mm = fmaxf(mm, s_red[w]);
        s_red[0] = mm;
    }
    __syncthreads();

    const float rowmax = s_red[0];
    const float thr = (rowmax < 2.0f) ? (rowmax - 1e-6f) : 2.0f;

    // ---- Stage 3: compact actives into LDS list --------------------------
    #pragma unroll
    for (int it = 0; it < CHUNKS; ++it) {
        const int i4 = tid + it * TPB;
        const float4 v = lrow[i4];
        const int base = i4 * 4;
        float e[4] = {v.x, v.y, v.z, v.w};
        #pragma unroll
        for (int c = 0; c < 4; ++c) {
            if (e[c] > thr) {
                const int pos = atomicAdd(&s_cnt, 1);   // ds_add_rtn_u32
                if (pos < CAP) { s_val[pos] = e[c]; s_idx[pos] = base + c; }
            }
        }
    }
    __syncthreads();

    const int cnt = (s_cnt < CAP) ? s_cnt : CAP;

    // pad tail with -inf sentinels so they sink in the descending sort
    for (int i = tid; i < CAP; i += TPB) {
        if (i >= cnt) { s_val[i] = -INFINITY; s_idx[i] = 0x7FFFFFFF; }
    }

    // ---- Stage 4: bitonic sort (descending) ------------------------------
    for (int k = 2; k <= CAP; k <<= 1) {
        for (int j = k >> 1; j > 0; j >>= 1) {
            __syncthreads();
            for (int i = tid; i < CAP; i += TPB) {
                const int ixj = i ^ j;
                if (ixj > i) {
                    const float a = s_val[i], b = s_val[ixj];
                    const bool sw = ((i & k) == 0) ? (a < b) : (a > b);
                    if (sw) {
                        s_val[i] = b; s_val[ixj] = a;
                        const int t = s_idx[i]; s_idx[i] = s_idx[ixj]; s_idx[ixj] = t;
                    }
                }
            }
        }
    }
    __syncthreads();

    // ---- Stage 5: write top-K (values then indices regions) --------------
    const size_t obase = (size_t)row * (size_t)K;
    for (int j = tid; j < K; j += TPB) {
        const bool act = (j < cnt);
        out_val[obase + j] = act ? s_val[j] : 0.0f;
        out_idx[obase + j] = act ? s_idx[j] : 0;
    }
}

extern "C" void kernel_launch(void* const* d_in, const int* in_sizes, int n_in,
                              void* d_out, int out_size, void* d_ws, size_t ws_size,
                              hipStream_t stream) {
    (void)n_in; (void)d_ws; (void)ws_size;
    const float* x = (const float*)d_in[0];
    const int rows = in_sizes[0] / NF;
    if (rows <= 0 || out_size <= 0) return;
    const int K = out_size / (2 * rows);   // out = [rows*K floats][rows*K ints]
    if (K <= 0) return;

    float* out_val = (float*)d_out;
    int*   out_idx = (int*)(out_val + (size_t)rows * (size_t)K);

    jumprelu_topk_kernel<<<dim3(rows), dim3(TPB), 0, stream>>>(x, out_val, out_idx, K);
}